// NamespaceAwareAttention_87582973100388
// MI455X (gfx1250) — compile-verified
//
#include <hip/hip_runtime.h>
#include <hip/hip_bf16.h>
#include <math.h>

// ---------------------------------------------------------------------------
// NamespaceAwareAttention on gfx1250 (MI455X), full-fp32 WMMA path.
//   X[2048,2048] -> Q = X@Wq [2048,2048], K = X@Wk [2048,512], V = X@Wv [2048,512]
//   per-head flash attention with trust-gap score mask
//   out = attn @ Wo
// All GEMM math runs on V_WMMA_F32_16X16X4_F32 (exact fp32 tensor op).
// GEMM strides are compile-time template parameters so inner-loop addressing
// strength-reduces to pointer bumps + immediate offsets; loads for step i+1
// are issued before step i's 8 back-to-back WMMAs (partial s_wait_loadcnt).
// ---------------------------------------------------------------------------

typedef float v2f __attribute__((ext_vector_type(2)));
typedef float v8f __attribute__((ext_vector_type(8)));

#define SEQ     2048
#define HID     2048
#define NHEADS  32
#define NKV     8
#define HDIM    64
#define KVN     (NKV * HDIM)     // 512
#define QN      (NHEADS * HDIM)  // 2048

__device__ __forceinline__ v8f v8f_zero() {
  v8f z = {0.f, 0.f, 0.f, 0.f, 0.f, 0.f, 0.f, 0.f};
  return z;
}

// D = A(16x4 fp32) * B(4x16 fp32) + C(16x16 fp32)
__device__ __forceinline__ v8f wmma4(v2f a, v2f b, v8f c) {
  return __builtin_amdgcn_wmma_f32_16x16x4_f32(
      /*neg_a=*/false, a, /*neg_b=*/false, b,
      /*c_mod=*/(short)0, c, /*reuse_a=*/false, /*reuse_b=*/false);
}

// ---------------------------------------------------------------------------
// fp32 WMMA GEMM: C[M,N] = A[M,K] @ B[K,N], row-major, N/K compile-time.
// One wave -> 32x64 output tile (2 M-subtiles x 4 N-subtiles = 8 accums).
// B fragments shared across both M-subtiles; register double-buffered K-loop
// with peeled epilogue (no per-iteration clamp/select).
// ---------------------------------------------------------------------------
template <int N, int K>
__global__ __launch_bounds__(256)
void gemm_wmma_f32(const float* __restrict__ A, const float* __restrict__ B,
                   float* __restrict__ C, int M) {
  const int wave  = (blockIdx.x * blockDim.x + threadIdx.x) >> 5;
  const int lane  = threadIdx.x & 31;
  const int n16   = lane & 15;
  const int half  = lane >> 4;
  constexpr int ntile = N >> 6;          // 64-wide column tiles
  const int tm    = wave / ntile;        // 32-row tiles
  const int tn    = wave % ntile;
  if (tm >= (M >> 5)) return;            // wave-uniform guard (EXEC stays full)

  v8f acc[2][4];
#pragma unroll
  for (int mi = 0; mi < 2; ++mi)
#pragma unroll
    for (int j = 0; j < 4; ++j) acc[mi][j] = v8f_zero();

  // Per-lane fragment pointers; all loop addressing is pointer bumps.
  const float* aP0 = A + (size_t)(tm * 32 + n16) * K + 2 * half;       // M-sub 0
  const float* aP1 = aP0 + (size_t)16 * K;                             // M-sub 1
  const float* bP  = B + (size_t)(2 * half) * N + tn * 64 + n16;

  // ---- prologue: K-step 0 fragments ----
  v2f a0c, a1c, bc[4];
  a0c.x = aP0[0]; a0c.y = aP0[1];
  a1c.x = aP1[0]; a1c.y = aP1[1];
#pragma unroll
  for (int j = 0; j < 4; ++j) { bc[j].x = bP[j * 16]; bc[j].y = bP[N + j * 16]; }

  // ---- steady state: prefetch next step, then 8 back-to-back WMMAs ----
  for (int it = 0; it < K / 4 - 1; ++it) {
    aP0 += 4; aP1 += 4; bP += (size_t)4 * N;
    v2f a0n, a1n, bn[4];
    a0n.x = aP0[0]; a0n.y = aP0[1];
    a1n.x = aP1[0]; a1n.y = aP1[1];
#pragma unroll
    for (int j = 0; j < 4; ++j) { bn[j].x = bP[j * 16]; bn[j].y = bP[N + j * 16]; }

#pragma unroll
    for (int j = 0; j < 4; ++j) acc[0][j] = wmma4(a0c, bc[j], acc[0][j]);
#pragma unroll
    for (int j = 0; j < 4; ++j) acc[1][j] = wmma4(a1c, bc[j], acc[1][j]);

    a0c = a0n; a1c = a1n;
#pragma unroll
    for (int j = 0; j < 4; ++j) bc[j] = bn[j];
  }

  // ---- epilogue: final K-step ----
#pragma unroll
  for (int j = 0; j < 4; ++j) acc[0][j] = wmma4(a0c, bc[j], acc[0][j]);
#pragma unroll
  for (int j = 0; j < 4; ++j) acc[1][j] = wmma4(a1c, bc[j], acc[1][j]);

  const int col0 = tn * 64;
#pragma unroll
  for (int mi = 0; mi < 2; ++mi)
#pragma unroll
    for (int j = 0; j < 4; ++j)
#pragma unroll
      for (int i = 0; i < 8; ++i)
        C[(size_t)(tm * 32 + mi * 16 + i + 8 * half) * N + col0 + j * 16 + n16] =
            acc[mi][j][i];
}

// ---------------------------------------------------------------------------
// Flash attention, one wave per (head, 16-query tile).
//   scores = (Q Kt)/8 * trustMask, online softmax, O += P@V
// Q fragments register-resident; K fragments batch-loaded (one wait, then 16
// back-to-back WMMAs); P restaged C-layout -> A-layout via 1KB per-wave LDS;
// V fragments double-buffered across the 4 P@V K-steps.
// ---------------------------------------------------------------------------
__global__ __launch_bounds__(128)
void attn_flash_wmma(const float* __restrict__ Q, const float* __restrict__ Km,
                     const float* __restrict__ Vm, const int* __restrict__ ns,
                     float* __restrict__ attnOut) {
  __shared__ float pbuf[4][16 * 16];

  const int widx = threadIdx.x >> 5;
  const int lane = threadIdx.x & 31;
  const int n16  = lane & 15;
  const int half = lane >> 4;

  const int tile = blockIdx.x * 4 + widx;     // 4096 tiles total
  const int h    = tile >> 7;                 // head 0..31
  const int t    = tile & 127;                // query tile 0..127
  const int q0   = t * 16;
  const int kvb  = (h >> 2) * HDIM;           // GQA: kv head = h/4
  const int qb   = h * HDIM;

  // Register-resident Q fragments: 16 K-steps covering HDIM=64
  v2f qf[16];
  {
    const float* qrow = Q + (size_t)(q0 + n16) * QN + qb + 2 * half;
#pragma unroll
    for (int ks = 0; ks < 16; ++ks) {
      qf[ks].x = qrow[ks * 4]; qf[ks].y = qrow[ks * 4 + 1];
    }
  }

  // Trust level of the 8 query rows this lane owns in C-layout
  int qtr[8];
#pragma unroll
  for (int i = 0; i < 8; ++i) qtr[i] = ns[q0 + i + 8 * half];

  float mrun[8], lrun[8];
#pragma unroll
  for (int i = 0; i < 8; ++i) { mrun[i] = -INFINITY; lrun[i] = 0.f; }

  v8f o[4];
  o[0] = v8f_zero(); o[1] = v8f_zero(); o[2] = v8f_zero(); o[3] = v8f_zero();

  for (int kt = 0; kt < SEQ / 16; ++kt) {
    const int key0 = kt * 16;

    // --- batch-load all 16 K fragments: B[k][n] = K[key0+n][kvb+k] ---
    v2f kf[16];
    {
      const float* kp = Km + (size_t)(key0 + n16) * KVN + kvb + 2 * half;
#pragma unroll
      for (int ks = 0; ks < 16; ++ks) {
        kf[ks].x = kp[ks * 4]; kf[ks].y = kp[ks * 4 + 1];
      }
    }
    // --- S = Q * K^T : 16 back-to-back WMMAs ---
    v8f s = v8f_zero();
#pragma unroll
    for (int ks = 0; ks < 16; ++ks) s = wmma4(qf[ks], kf[ks], s);

    const int ktr = ns[key0 + n16];    // key trust for this lane's column

    // --- scale + trust mask + online softmax (row reductions across 16 lanes)
    float alpha[8];
#pragma unroll
    for (int i = 0; i < 8; ++i) {
      const int gap = qtr[i] - ktr;
      const float msk = (gap >= 60) ? 0.5f : ((gap >= 30) ? 0.7f : 1.0f);
      float sv = s[i] * 0.125f * msk;

      float rmax = sv;
#pragma unroll
      for (int off = 1; off < 16; off <<= 1)
        rmax = fmaxf(rmax, __shfl_xor(rmax, off, 16));

      const float mnew = fmaxf(mrun[i], rmax);
      const float p = __expf(sv - mnew);

      float rsum = p;
#pragma unroll
      for (int off = 1; off < 16; off <<= 1)
        rsum += __shfl_xor(rsum, off, 16);

      alpha[i]  = __expf(mrun[i] - mnew);
      lrun[i]   = lrun[i] * alpha[i] + rsum;
      mrun[i]   = mnew;
      s[i]      = p;                   // s now holds P tile (C-layout)
    }

    // rescale running O by alpha (row-matched: index i <-> row i+8*half)
#pragma unroll
    for (int j = 0; j < 4; ++j)
#pragma unroll
      for (int i = 0; i < 8; ++i) o[j][i] *= alpha[i];

    // --- restage P: C-layout -> LDS -> A-layout ---
#pragma unroll
    for (int i = 0; i < 8; ++i)
      pbuf[widx][(i + 8 * half) * 16 + n16] = s[i];
    __syncthreads();

    // --- O += P(16x16) @ V(16x64), V fragments double-buffered ---
    v2f pf[4];
#pragma unroll
    for (int ks = 0; ks < 4; ++ks) {
      const int d = ks * 4 + 2 * half;
      pf[ks].x = pbuf[widx][n16 * 16 + d];
      pf[ks].y = pbuf[widx][n16 * 16 + d + 1];
    }
    v2f vcur[4], vnxt[4];
    {
      const float* v0 = Vm + (size_t)(key0 + 2 * half) * KVN + kvb + n16;
#pragma unroll
      for (int j = 0; j < 4; ++j) {
        vcur[j].x = v0[j * 16]; vcur[j].y = v0[KVN + j * 16];
      }
    }
#pragma unroll
    for (int ks = 0; ks < 4; ++ks) {
      if (ks < 3) {
        const float* v0 =
            Vm + (size_t)(key0 + (ks + 1) * 4 + 2 * half) * KVN + kvb + n16;
#pragma unroll
        for (int j = 0; j < 4; ++j) {
          vnxt[j].x = v0[j * 16]; vnxt[j].y = v0[KVN + j * 16];
        }
      }
#pragma unroll
      for (int j = 0; j < 4; ++j) o[j] = wmma4(pf[ks], vcur[j], o[j]);
#pragma unroll
      for (int j = 0; j < 4; ++j) vcur[j] = vnxt[j];
    }
    __syncthreads();
  }

  // --- normalize and write attn in [S, H] layout (head h at cols h*64..) ---
  float inv[8];
#pragma unroll
  for (int i = 0; i < 8; ++i) inv[i] = 1.0f / lrun[i];
#pragma unroll
  for (int j = 0; j < 4; ++j)
#pragma unroll
    for (int i = 0; i < 8; ++i)
      attnOut[(size_t)(q0 + i + 8 * half) * QN + qb + j * 16 + n16] =
          o[j][i] * inv[i];
}

// ---------------------------------------------------------------------------
extern "C" void kernel_launch(void* const* d_in, const int* in_sizes, int n_in,
                              void* d_out, int out_size, void* d_ws, size_t ws_size,
                              hipStream_t stream) {
  const float* X  = (const float*)d_in[0];   // [1,2048,2048]
  const int*   ns = (const int*)  d_in[1];   // [1,2048]
  const float* Wq = (const float*)d_in[2];   // [2048,2048]
  const float* Wk = (const float*)d_in[3];   // [2048,512]
  const float* Wv = (const float*)d_in[4];   // [2048,512]
  const float* Wo = (const float*)d_in[5];   // [2048,2048]
  float* out = (float*)d_out;                // [1,2048,2048]

  float* Qw  = (float*)d_ws;                 // 2048*2048
  float* Kw  = Qw + (size_t)SEQ * QN;        // 2048*512
  float* Vw  = Kw + (size_t)SEQ * KVN;       // 2048*512
  float* At  = Vw + (size_t)SEQ * KVN;       // 2048*2048

  // QKV projections: waves = (M/32)*(N/64); 8 waves (256 thr) per block
  gemm_wmma_f32<QN,  HID><<<(64 * 32) / 8, 256, 0, stream>>>(X, Wq, Qw, SEQ);
  gemm_wmma_f32<KVN, HID><<<(64 *  8) / 8, 256, 0, stream>>>(X, Wk, Kw, SEQ);
  gemm_wmma_f32<KVN, HID><<<(64 *  8) / 8, 256, 0, stream>>>(X, Wv, Vw, SEQ);

  // Flash attention: 32 heads x 128 q-tiles = 4096 waves, 4 waves/block
  attn_flash_wmma<<<4096 / 4, 128, 0, stream>>>(Qw, Kw, Vw, ns, At);

  // Output projection
  gemm_wmma_f32<HID, HID><<<(64 * 32) / 8, 256, 0, stream>>>(At, Wo, out, SEQ);
}